// TransformerDDPMRegNet_34308198761278
// MI455X (gfx1250) — compile-verified
//
#include <hip/hip_runtime.h>
#include <hip/hip_bf16.h>

// ---------------------------------------------------------------------------
// Point Transformer (4 layers) + DDPM head for MI455X (gfx1250, wave32).
// Dense [4096,128]@[128,128] GEMMs use v_wmma_f32_16x16x4_f32 (fp32 WMMA).
// Algebraic simplifications: Wq unused; K-proj reduced to rowsum matvec;
// pe matmul over K collapsed to one post-aggregation GEMM.
// ---------------------------------------------------------------------------

typedef float v2f __attribute__((ext_vector_type(2)));
typedef float v8f __attribute__((ext_vector_type(8)));

#define NROWS 4096      // B*N
#define NPTS  2048      // N per batch
#define DD    128
#define KNN   128

// ----------------------------- kNN (once) ----------------------------------
// One block per point: compute 2048 distances (same sq-formula as reference),
// bitonic-sort packed (orderable-dist-bits << 32 | idx) keys in LDS, emit the
// 128 smallest (matches top_k(-dist) tie-breaking by index).
__global__ __launch_bounds__(256) void knn_kernel(const float* __restrict__ xyz,
                                                  int* __restrict__ knn) {
  __shared__ unsigned long long key[NPTS];
  int p = blockIdx.x;                 // global point row
  int b = p >> 11;
  int base = b << 11;
  float xi = xyz[p * 3 + 0], yi = xyz[p * 3 + 1], zi = xyz[p * 3 + 2];
  float sqi = xi * xi + yi * yi + zi * zi;
  for (int j = threadIdx.x; j < NPTS; j += 256) {
    int gj = base + j;
    float xj = xyz[gj * 3 + 0], yj = xyz[gj * 3 + 1], zj = xyz[gj * 3 + 2];
    float sqj = xj * xj + yj * yj + zj * zj;
    float d = sqi + sqj - 2.0f * (xi * xj + yi * yj + zi * zj);
    unsigned int bits = __float_as_uint(d);
    bits = (bits & 0x80000000u) ? ~bits : (bits | 0x80000000u);  // order-preserving map
    key[j] = ((unsigned long long)bits << 32) | (unsigned int)j;
  }
  __syncthreads();
  for (int k2 = 2; k2 <= NPTS; k2 <<= 1) {
    for (int j = k2 >> 1; j > 0; j >>= 1) {
      for (int i = threadIdx.x; i < NPTS; i += 256) {
        int ixj = i ^ j;
        if (ixj > i) {
          bool up = ((i & k2) == 0);
          unsigned long long a = key[i], c = key[ixj];
          if (up ? (a > c) : (a < c)) { key[i] = c; key[ixj] = a; }
        }
      }
      __syncthreads();
    }
  }
  if (threadIdx.x < KNN)
    knn[p * KNN + threadIdx.x] = base + (int)(key[threadIdx.x] & 0xFFFFFFFFu);
}

// ------------------------ tiny per-layer helpers ---------------------------
__global__ __launch_bounds__(128) void rowsum2_kernel(const float* __restrict__ Wk,
                                                      const float* __restrict__ Wp2,
                                                      float* __restrict__ wkrs,
                                                      float* __restrict__ w2s) {
  int j = threadIdx.x;
  float s1 = 0.f, s2 = 0.f;
  for (int d = 0; d < DD; ++d) { s1 += Wk[j * DD + d]; s2 += Wp2[j * DD + d]; }
  wkrs[j] = s1;
  w2s[j] = s2;
}

__global__ __launch_bounds__(256) void sk_kernel(const float* __restrict__ fused,
                                                 const float* __restrict__ wkrs,
                                                 float* __restrict__ sk) {
  __shared__ float w[DD];
  if (threadIdx.x < DD) w[threadIdx.x] = wkrs[threadIdx.x];
  __syncthreads();
  int r = blockIdx.x * 256 + threadIdx.x;
  const float* fr = fused + r * DD;
  float s = 0.f;
  for (int j = 0; j < DD; ++j) s += fr[j] * w[j];
  sk[r] = s;
}

// --------------------------- WMMA f32 GEMM ---------------------------------
// C[4096,128] = A[4096,128] @ W[128,128] + bias, with epilogues:
//  mode 0: plain    mode 1: + Add[m,n]    mode 2: Res[m,n] + gamma*(...)
// One wave per 16x16 tile; K stepped by 4 via v_wmma_f32_16x16x4_f32.
__global__ __launch_bounds__(128) void gemm128_kernel(const float* __restrict__ A,
                                                      const float* __restrict__ W,
                                                      const float* __restrict__ bias,
                                                      float* __restrict__ C,
                                                      const float* __restrict__ aux,
                                                      const float* __restrict__ gamma_ptr,
                                                      int mode) {
  int wave = threadIdx.x >> 5;
  int lane = threadIdx.x & 31;
  int tile = blockIdx.x * 4 + wave;     // 2048 tiles total
  int mtile = tile >> 3;                // 256 M tiles
  int ntile = tile & 7;                 // 8 N tiles
  int hi = lane >> 4;
  int lo = lane & 15;
  int mrow = mtile * 16 + lo;
  int ncol = ntile * 16 + lo;
  const float* arow = A + mrow * DD;
  v8f acc = {0.f, 0.f, 0.f, 0.f, 0.f, 0.f, 0.f, 0.f};
  for (int kk = 0; kk < DD; kk += 4) {
    int ka = kk + 2 * hi;
    v2f a, b;
    a.x = arow[ka];
    a.y = arow[ka + 1];
    b.x = W[ka * DD + ncol];
    b.y = W[(ka + 1) * DD + ncol];
    acc = __builtin_amdgcn_wmma_f32_16x16x4_f32(false, a, false, b, (short)0, acc,
                                                false, false);
  }
  float g = (mode == 2) ? gamma_ptr[0] : 0.0f;
#pragma unroll
  for (int r = 0; r < 8; ++r) {
    int m = mtile * 16 + r + 8 * hi;
    float c = acc[r] + bias[ncol];
    if (mode == 1) c += aux[m * DD + ncol];
    else if (mode == 2) c = aux[m * DD + ncol] + g * c;
    C[m * DD + ncol] = c;
  }
}

// ------------------------- per-point attention -----------------------------
// 128 threads = K neighbors (pass 1: logits) then = D channels (pass 2).
// h = relu(rel@Wp1+b1) recomputed on the fly both passes (cheap: 3 FMAs each).
__global__ __launch_bounds__(128) void attn_kernel(const float* __restrict__ xyz,
                                                   const int* __restrict__ knn,
                                                   const float* __restrict__ sk,
                                                   const float* __restrict__ Vm,
                                                   const float* __restrict__ Wp1,
                                                   const float* __restrict__ bp1,
                                                   const float* __restrict__ w2s,
                                                   float* __restrict__ hbar,
                                                   float* __restrict__ vbar) {
  __shared__ float relS[KNN][3];
  __shared__ int nbrS[KNN];
  __shared__ float attS[KNN];
  int n = blockIdx.x;
  int t = threadIdx.x;
  int nbr = knn[n * KNN + t];
  nbrS[t] = nbr;
  float rx = xyz[n * 3 + 0] - xyz[nbr * 3 + 0];
  float ry = xyz[n * 3 + 1] - xyz[nbr * 3 + 1];
  float rz = xyz[n * 3 + 2] - xyz[nbr * 3 + 2];
  relS[t][0] = rx; relS[t][1] = ry; relS[t][2] = rz;
  float logit = -sk[nbr];
  for (int d = 0; d < DD; ++d) {
    float h = rx * Wp1[d] + ry * Wp1[DD + d] + rz * Wp1[2 * DD + d] + bp1[d];
    logit += fmaxf(h, 0.f) * w2s[d];
  }
  attS[t] = logit;
  __syncthreads();
  float mx = -1e30f;
  for (int k = 0; k < KNN; ++k) mx = fmaxf(mx, attS[k]);
  __syncthreads();
  float e = expf(logit - mx);
  attS[t] = e;
  __syncthreads();
  float sum = 0.f;
  for (int k = 0; k < KNN; ++k) sum += attS[k];
  float inv = 1.0f / sum;
  // pass 2: t is now a channel index
  float w0 = Wp1[t], w1 = Wp1[DD + t], w2c = Wp1[2 * DD + t], bb = bp1[t];
  float hb = 0.f, vb = 0.f;
  for (int k = 0; k < KNN; ++k) {
    float a = attS[k];
    float h = relS[k][0] * w0 + relS[k][1] * w1 + relS[k][2] * w2c + bb;
    hb += a * fmaxf(h, 0.f);
    vb += a * Vm[nbrS[k] * DD + t];
  }
  hbar[n * DD + t] = hb * inv;
  vbar[n * DD + t] = vb * inv;
}

// --------------------------- DDPM head pieces ------------------------------
__global__ __launch_bounds__(128) void temb_kernel(const int* __restrict__ t,
                                                   const float* __restrict__ tW1,
                                                   const float* __restrict__ tb1,
                                                   const float* __restrict__ tW2,
                                                   const float* __restrict__ tb2,
                                                   float* __restrict__ temb2) {
  __shared__ float emb[DD];
  __shared__ float h1[DD];
  int b = blockIdx.x, d = threadIdx.x;
  float tv = (float)t[b];
  int i = (d < 64) ? d : d - 64;
  float freq = expf(-logf(10000.0f) * (float)i / 63.0f);
  float arg = tv * freq;
  emb[d] = (d < 64) ? sinf(arg) : cosf(arg);
  __syncthreads();
  float s = tb1[d];
  for (int j = 0; j < DD; ++j) s += emb[j] * tW1[j * DD + d];
  h1[d] = fmaxf(s, 0.f);
  __syncthreads();
  float s2 = tb2[d];
  for (int j = 0; j < DD; ++j) s2 += h1[j] * tW2[j * DD + d];
  temb2[b * DD + d] = s2;
}

// H1 = concat(x_t[3], fused[128], temb[128]) @ nW1[259,128] + nb1, via WMMA
// for the two 128-wide slabs + 3 scalar FMAs for x_t.
__global__ __launch_bounds__(128) void headgemm1_kernel(const float* __restrict__ xt,
                                                        const float* __restrict__ fused,
                                                        const float* __restrict__ temb,
                                                        const float* __restrict__ W,
                                                        const float* __restrict__ bias,
                                                        float* __restrict__ C) {
  int wave = threadIdx.x >> 5;
  int lane = threadIdx.x & 31;
  int tile = blockIdx.x * 4 + wave;
  int mtile = tile >> 3;
  int ntile = tile & 7;
  int hi = lane >> 4;
  int lo = lane & 15;
  int mrow = mtile * 16 + lo;
  int ncol = ntile * 16 + lo;
  v8f acc = {0.f, 0.f, 0.f, 0.f, 0.f, 0.f, 0.f, 0.f};
  const float* frow = fused + mrow * DD;
  for (int kk = 0; kk < DD; kk += 4) {
    int ka = kk + 2 * hi;
    v2f a, b;
    a.x = frow[ka];
    a.y = frow[ka + 1];
    b.x = W[(3 + ka) * DD + ncol];
    b.y = W[(4 + ka) * DD + ncol];
    acc = __builtin_amdgcn_wmma_f32_16x16x4_f32(false, a, false, b, (short)0, acc,
                                                false, false);
  }
  int bidx = (mtile * 16) >> 11;  // batch is uniform within a 16-row tile
  const float* trow = temb + bidx * DD;
  for (int kk = 0; kk < DD; kk += 4) {
    int ka = kk + 2 * hi;
    v2f a, b;
    a.x = trow[ka];
    a.y = trow[ka + 1];
    b.x = W[(131 + ka) * DD + ncol];
    b.y = W[(132 + ka) * DD + ncol];
    acc = __builtin_amdgcn_wmma_f32_16x16x4_f32(false, a, false, b, (short)0, acc,
                                                false, false);
  }
  float w0 = W[ncol], w1 = W[DD + ncol], w2c = W[2 * DD + ncol];
#pragma unroll
  for (int r = 0; r < 8; ++r) {
    int m = mtile * 16 + r + 8 * hi;
    float c = acc[r] + bias[ncol] + xt[m * 3 + 0] * w0 + xt[m * 3 + 1] * w1 +
              xt[m * 3 + 2] * w2c;
    C[m * DD + ncol] = c;
  }
}

__global__ __launch_bounds__(128) void ln_relu_kernel(const float* __restrict__ X,
                                                      const float* __restrict__ g,
                                                      const float* __restrict__ beta,
                                                      float* __restrict__ Y) {
  __shared__ float xs[DD];
  int r = blockIdx.x, t = threadIdx.x;
  float v = X[r * DD + t];
  xs[t] = v;
  __syncthreads();
  float m = 0.f;
  for (int k = 0; k < DD; ++k) m += xs[k];
  m *= (1.0f / DD);
  float var = 0.f;
  for (int k = 0; k < DD; ++k) { float dd = xs[k] - m; var += dd * dd; }
  var *= (1.0f / DD);
  float y = (v - m) * rsqrtf(var + 1e-5f) * g[t] + beta[t];
  Y[r * DD + t] = fmaxf(y, 0.f);
}

__global__ __launch_bounds__(256) void out_kernel(const float* __restrict__ h2,
                                                  const float* __restrict__ nW3,
                                                  const float* __restrict__ nb3,
                                                  float* __restrict__ out) {
  int gid = blockIdx.x * 256 + threadIdx.x;
  if (gid >= NROWS * 3) return;
  int r = gid / 3, c = gid % 3;
  float s = nb3[c];
  for (int j = 0; j < DD; ++j) s += h2[r * DD + j] * nW3[j * 3 + c];
  out[gid] = s;
}

// ------------------------------- launcher ----------------------------------
extern "C" void kernel_launch(void* const* d_in, const int* in_sizes, int n_in,
                              void* d_out, int out_size, void* d_ws, size_t ws_size,
                              hipStream_t stream) {
  const float* xyz  = (const float*)d_in[0];
  const float* feat = (const float*)d_in[1];
  const float* xt   = (const float*)d_in[2];
  const int*   tt   = (const int*)d_in[3];
  // d_in[4]/[5] (Wq/bq) provably unused: their contribution cancels in softmax.
  const float* Wk    = (const float*)d_in[6];
  const float* Wv    = (const float*)d_in[8];
  const float* bv    = (const float*)d_in[9];
  const float* Wp1   = (const float*)d_in[10];
  const float* bp1   = (const float*)d_in[11];
  const float* Wp2   = (const float*)d_in[12];
  const float* bp2   = (const float*)d_in[13];
  const float* gamma = (const float*)d_in[14];
  const float* Wpr   = (const float*)d_in[15];
  const float* bpr   = (const float*)d_in[16];
  const float* tW1 = (const float*)d_in[17];
  const float* tb1 = (const float*)d_in[18];
  const float* tW2 = (const float*)d_in[19];
  const float* tb2 = (const float*)d_in[20];
  const float* nW1 = (const float*)d_in[21];
  const float* nb1 = (const float*)d_in[22];
  const float* ng1 = (const float*)d_in[23];
  const float* nbe1 = (const float*)d_in[24];
  const float* nW2 = (const float*)d_in[25];
  const float* nb2 = (const float*)d_in[26];
  const float* ng2 = (const float*)d_in[27];
  const float* nbe2 = (const float*)d_in[28];
  const float* nW3 = (const float*)d_in[29];
  const float* nb3 = (const float*)d_in[30];

  // workspace carve (all 256B aligned)
  char* ws = (char*)d_ws;
  size_t off = 0;
  auto carve = [&](size_t bytes) {
    void* p = ws + off;
    off += (bytes + 255) & ~(size_t)255;
    return p;
  };
  int*   knn    = (int*)  carve(NROWS * KNN * sizeof(int));
  float* skbuf  = (float*)carve(NROWS * sizeof(float));
  float* wkrs   = (float*)carve(DD * sizeof(float));
  float* w2s    = (float*)carve(DD * sizeof(float));
  float* temb2  = (float*)carve(2 * DD * sizeof(float));
  float* fusedA = (float*)carve(NROWS * DD * sizeof(float));
  float* fusedB = (float*)carve(NROWS * DD * sizeof(float));
  float* Vm     = (float*)carve(NROWS * DD * sizeof(float));
  float* hbar   = (float*)carve(NROWS * DD * sizeof(float));
  float* vbar   = (float*)carve(NROWS * DD * sizeof(float));
  float* agg    = (float*)carve(NROWS * DD * sizeof(float));
  float* hb1    = (float*)carve(NROWS * DD * sizeof(float));
  float* hb1n   = (float*)carve(NROWS * DD * sizeof(float));
  float* hb2    = (float*)carve(NROWS * DD * sizeof(float));
  float* hb2n   = (float*)carve(NROWS * DD * sizeof(float));
  (void)ws_size; (void)in_sizes; (void)n_in; (void)out_size;

  // kNN once: xyz is constant across layers
  knn_kernel<<<NROWS, 256, 0, stream>>>(xyz, knn);

  const float* cur = feat;
  float* bufs[2] = {fusedA, fusedB};
  for (int L = 0; L < 4; ++L) {
    const float* WkL = Wk + (size_t)L * DD * DD;
    const float* WvL = Wv + (size_t)L * DD * DD;
    const float* bvL = bv + (size_t)L * DD;
    const float* Wp1L = Wp1 + (size_t)L * 3 * DD;
    const float* bp1L = bp1 + (size_t)L * DD;
    const float* Wp2L = Wp2 + (size_t)L * DD * DD;
    const float* bp2L = bp2 + (size_t)L * DD;
    const float* WprL = Wpr + (size_t)L * DD * DD;
    const float* bprL = bpr + (size_t)L * DD;

    rowsum2_kernel<<<1, 128, 0, stream>>>(WkL, Wp2L, wkrs, w2s);
    sk_kernel<<<NROWS / 256, 256, 0, stream>>>(cur, wkrs, skbuf);
    gemm128_kernel<<<512, 128, 0, stream>>>(cur, WvL, bvL, Vm, nullptr, nullptr, 0);
    attn_kernel<<<NROWS, 128, 0, stream>>>(xyz, knn, skbuf, Vm, Wp1L, bp1L, w2s,
                                           hbar, vbar);
    gemm128_kernel<<<512, 128, 0, stream>>>(hbar, Wp2L, bp2L, agg, vbar, nullptr, 1);
    float* nxt = bufs[L & 1];
    gemm128_kernel<<<512, 128, 0, stream>>>(agg, WprL, bprL, nxt, cur, gamma + L, 2);
    cur = nxt;
  }

  // DDPM head
  temb_kernel<<<2, 128, 0, stream>>>(tt, tW1, tb1, tW2, tb2, temb2);
  headgemm1_kernel<<<512, 128, 0, stream>>>(xt, cur, temb2, nW1, nb1, hb1);
  ln_relu_kernel<<<NROWS, 128, 0, stream>>>(hb1, ng1, nbe1, hb1n);
  gemm128_kernel<<<512, 128, 0, stream>>>(hb1n, nW2, nb2, hb2, nullptr, nullptr, 0);
  ln_relu_kernel<<<NROWS, 128, 0, stream>>>(hb2, ng2, nbe2, hb2n);
  out_kernel<<<(NROWS * 3 + 255) / 256, 256, 0, stream>>>(hb2n, nW3, nb3,
                                                          (float*)d_out);
}